// GraphMultiHeadAttentionLayer_44513041056580
// MI455X (gfx1250) — compile-verified
//
#include <hip/hip_runtime.h>
#include <math.h>

// Problem constants (fixed by the reference harness)
#define FN 64
#define FE 16
#define DIM 128
#define NHEAD 4
#define HD (NHEAD * DIM) // 512
#define LS 132           // padded LDS row stride for 16x128 tiles (bank spread)
#define ES 20            // padded LDS row stride for 16x16 edge tile

typedef __attribute__((ext_vector_type(2))) float v2f;
typedef __attribute__((ext_vector_type(8))) float v8f;

// ---- FP32 WMMA 16x16x4 (exact fp32 math, matches reference precision) ----
__device__ __forceinline__ v8f wmma4(v2f a, v2f b, v8f c) {
  return __builtin_amdgcn_wmma_f32_16x16x4_f32(false, a, false, b, (short)0, c,
                                               false, false);
}

// A fragment: 16x4 tile of a row-major matrix. Lanes 0-15: K=k0,k0+1 for row
// M=lane. Lanes 16-31: K=k0+2,k0+3 for row M=lane-16.
__device__ __forceinline__ v2f lda_frag(const float* p, int lda, int row0,
                                        int k0, int lane) {
  int half = lane >> 4, r = lane & 15;
  const float* q = p + (size_t)(row0 + r) * lda + k0 + 2 * half;
  v2f a;
  a.x = q[0];
  a.y = q[1];
  return a;
}

// B fragment from the packed weight layout: one b64 load per fragment.
// packed[(kstep*N + col)*2 + half] = {W[4k+2h][col], W[4k+2h+1][col]}
__device__ __forceinline__ v2f ldb_pk(const float2* p, int N, int k0, int col0,
                                      int lane) {
  int half = lane >> 4, c = lane & 15;
  float2 t = p[((size_t)(k0 >> 2) * N + col0 + c) * 2 + half];
  v2f b;
  b.x = t.x;
  b.y = t.y;
  return b;
}

// Async copy of 16 contiguous bytes global -> LDS (ASYNCcnt tracked).
__device__ __forceinline__ void async_load16(const float* gsrc, float* ldst) {
  unsigned l = (unsigned)(uintptr_t)ldst;              // LDS byte offset
  unsigned long long g = (unsigned long long)(uintptr_t)gsrc;
  asm volatile("global_load_async_to_lds_b128 %0, %1, off" ::"v"(l), "v"(g)
               : "memory");
}
__device__ __forceinline__ void wait_async0() {
  asm volatile("s_wait_asynccnt 0x0" ::: "memory");
}

__device__ __forceinline__ void atomicMaxF32(float* addr, float val) {
  int* ai = (int*)addr;
  int old = __float_as_int(*addr);
  while (__int_as_float(old) < val) {
    int assumed = old;
    old = atomicCAS(ai, assumed, __float_as_int(val));
    if (old == assumed) break;
  }
}

// ---------------- weight repack: row-major [K,N] -> B-fragment float2 ----------------
__global__ void k_pack(const float* __restrict__ src, float2* __restrict__ dst,
                       int K, int N) {
  int i = blockIdx.x * blockDim.x + threadIdx.x;
  int total = (K >> 2) * N * 2;
  if (i >= total) return;
  int kstep = i / (2 * N);
  int rem = i - kstep * 2 * N;
  int col = rem >> 1;
  int half = rem & 1;
  int k = 4 * kstep + 2 * half;
  float2 v;
  v.x = src[(size_t)k * N + col];
  v.y = src[(size_t)(k + 1) * N + col];
  dst[i] = v;
}

// ---------------- init: agg=0, segsum=0, segmax=-FLT_MAX ----------------
__global__ void k_init(float* __restrict__ segmax, float* __restrict__ segsum,
                       float* __restrict__ agg, int NH, int NHD) {
  int i = blockIdx.x * blockDim.x + threadIdx.x;
  if (i < NHD) agg[i] = 0.f;
  if (i < NH) {
    segmax[i] = -3.4028235e38f;
    segsum[i] = 0.f;
  }
}

// ---------------- x = nodes @ W_node + b_node ----------------
__global__ void __launch_bounds__(32)
k_proj_nodes(const float* __restrict__ nodes, const float2* __restrict__ pkWn,
             const float* __restrict__ bn, float* __restrict__ x, int N) {
  int tile = blockIdx.x;
  int tm = tile >> 3, tn = tile & 7;
  int lane = threadIdx.x;
  int row0 = tm * 16, col0 = tn * 16;
  if (row0 >= N) return;
  v8f acc = {};
#pragma unroll
  for (int k = 0; k < FN; k += 4)
    acc = wmma4(lda_frag(nodes, FN, row0, k, lane),
                ldb_pk(pkWn, DIM, k, col0, lane), acc);
  int half = lane >> 4, c = lane & 15;
  float bias = bn[col0 + c];
#pragma unroll
  for (int r = 0; r < 8; r++)
    x[(size_t)(row0 + r + 8 * half) * DIM + col0 + c] = acc[r] + bias;
}

// ---------------- per-edge-tile logits + segment max ----------------
// 128 threads = 4 waves per 16-edge tile; wave h handles head h.
__global__ void __launch_bounds__(128)
k_logits(const float* __restrict__ x, const float* __restrict__ edges,
         const int* __restrict__ senders, const int* __restrict__ receivers,
         const float2* __restrict__ pkWe, const float* __restrict__ be,
         const float2* __restrict__ pkWks, const float* __restrict__ bks,
         const float2* __restrict__ pkWkr, const float* __restrict__ bkr,
         float* __restrict__ wlog, float* __restrict__ segmax, int E) {
  __shared__ float sa[16 * LS];
  __shared__ float ra[16 * LS];
  __shared__ float ef[16 * LS];
  __shared__ float esh[16 * ES];
  int e0 = blockIdx.x * 16;
  int t = threadIdx.x;
  int lane = t & 31;
  int wave = t >> 5;

  // Async-stage the contiguous 16x16 edge-feature tile into LDS (64 x 16B).
  if (t < 64) {
    int row = t >> 2, cpos = (t & 3) * 4;
    async_load16(edges + (size_t)(e0 + row) * FE + cpos,
                 esh + row * ES + cpos);
  }
  // Prefetch next tile's edge block toward L2 while we work on this one.
  if (t == 64 && e0 + 16 < E) __builtin_prefetch(edges + (size_t)(e0 + 16) * FE, 0, 0);

  // Gather sa/ra tiles from L2-resident x: thread t loads 16 floats.
  {
    int row = t >> 3;
    int c0 = (t & 7) * 16;
    int e = min(e0 + row, E - 1);
    int s = senders[e];
    int r = receivers[e];
    const float4* xs = (const float4*)(x + (size_t)s * DIM + c0);
    const float4* xr = (const float4*)(x + (size_t)r * DIM + c0);
    float4* ds = (float4*)(sa + row * LS + c0);
    float4* dr = (float4*)(ra + row * LS + c0);
#pragma unroll
    for (int j = 0; j < 4; j++) {
      ds[j] = xs[j];
      dr[j] = xr[j];
    }
  }
  wait_async0();
  __syncthreads();

  // ef tile recomputed via K=16 WMMA from the LDS-staged edge tile.
#pragma unroll
  for (int bi = 0; bi < 2; bi++) {
    int nb = wave + 4 * bi;
    int col0 = nb * 16;
    v8f acc = {};
#pragma unroll
    for (int k = 0; k < FE; k += 4)
      acc = wmma4(lda_frag(esh, ES, 0, k, lane),
                  ldb_pk(pkWe, DIM, k, col0, lane), acc);
    int half = lane >> 4, c = lane & 15;
    float bias = be[col0 + c];
#pragma unroll
    for (int r = 0; r < 8; r++)
      ef[(r + 8 * half) * LS + col0 + c] = acc[r] + bias;
  }
  __syncthreads();

  // Hoist sa/ra A-fragments into registers once (reused across all 8 n-blocks).
  v2f asf[32], arf[32];
#pragma unroll
  for (int ks = 0; ks < 32; ks++) {
    asf[ks] = lda_frag(sa, LS, 0, 4 * ks, lane);
    arf[ks] = lda_frag(ra, LS, 0, 4 * ks, lane);
  }

  int h = wave;
  const float2* Ws = pkWks + (size_t)h * DIM * DIM / 2;
  const float2* Wr = pkWkr + (size_t)h * DIM * DIM / 2;
  int half = lane >> 4, c = lane & 15;
  float dot[8];
#pragma unroll
  for (int r = 0; r < 8; r++) dot[r] = 0.f;

  for (int nb = 0; nb < 8; nb++) {
    int col0 = nb * 16;
    v8f ks = {}, kr = {};
#pragma unroll
    for (int kk = 0; kk < 32; kk++) {
      ks = wmma4(asf[kk], ldb_pk(Ws, DIM, 4 * kk, col0, lane), ks);
      kr = wmma4(arf[kk], ldb_pk(Wr, DIM, 4 * kk, col0, lane), kr);
    }
    float bs = bks[h * DIM + col0 + c];
    float br = bkr[h * DIM + col0 + c];
#pragma unroll
    for (int r = 0; r < 8; r++) {
      float vks = ks[r] + bs;
      float vkr = kr[r] + br + ef[(r + 8 * half) * LS + col0 + c];
      float p = vks * vkr;
      // butterfly sum over the 16 N-columns held in this lane half
      p += __shfl_xor(p, 1);
      p += __shfl_xor(p, 2);
      p += __shfl_xor(p, 4);
      p += __shfl_xor(p, 8);
      dot[r] += p;
    }
  }
  const float scale = 0.08838834764831845f; // 1/sqrt(128)
  if ((lane & 15) == 0) {
#pragma unroll
    for (int r = 0; r < 8; r++) {
      int e = e0 + r + 8 * half;
      if (e < E) {
        float lg = dot[r] * scale;
        wlog[(size_t)h * E + e] = lg;
        atomicMaxF32(&segmax[receivers[e] * NHEAD + h], lg);
      }
    }
  }
}

// ---------------- w = exp(logit - segmax); segsum += w ----------------
__global__ void k_expsum(const float* __restrict__ segmax,
                         const int* __restrict__ receivers,
                         float* __restrict__ wlog, float* __restrict__ segsum,
                         int E) {
  int e = blockIdx.x * blockDim.x + threadIdx.x;
  int h = blockIdx.y;
  if (e >= E) return;
  int rc = receivers[e];
  float m = segmax[rc * NHEAD + h];
  float w = expf(wlog[(size_t)h * E + e] - m);
  wlog[(size_t)h * E + e] = w;
  atomicAdd(&segsum[rc * NHEAD + h], w);
}

// ---------------- scatter: agg[recv, h*D:] += (w/sum) * (sa + ef) ----------------
__global__ void __launch_bounds__(128)
k_msg(const float* __restrict__ x, const float* __restrict__ edges,
      const int* __restrict__ senders, const int* __restrict__ receivers,
      const float2* __restrict__ pkWe, const float* __restrict__ be,
      const float* __restrict__ wlog, const float* __restrict__ segsum,
      float* __restrict__ agg, int E) {
  __shared__ float sa[16 * LS];
  __shared__ float ef[16 * LS];
  __shared__ float esh[16 * ES];
  int e0 = blockIdx.x * 16;
  int t = threadIdx.x, lane = t & 31, wave = t >> 5;

  if (t < 64) {
    int row = t >> 2, cpos = (t & 3) * 4;
    async_load16(edges + (size_t)(e0 + row) * FE + cpos,
                 esh + row * ES + cpos);
  }
  if (t == 64 && e0 + 16 < E) __builtin_prefetch(edges + (size_t)(e0 + 16) * FE, 0, 0);

  {
    int row = t >> 3;
    int c0 = (t & 7) * 16;
    int s = senders[min(e0 + row, E - 1)];
    const float4* xs = (const float4*)(x + (size_t)s * DIM + c0);
    float4* ds = (float4*)(sa + row * LS + c0);
#pragma unroll
    for (int j = 0; j < 4; j++) ds[j] = xs[j];
  }
  wait_async0();
  __syncthreads();

#pragma unroll
  for (int bi = 0; bi < 2; bi++) {
    int nb = wave + 4 * bi, col0 = nb * 16;
    v8f acc = {};
#pragma unroll
    for (int k = 0; k < FE; k += 4)
      acc = wmma4(lda_frag(esh, ES, 0, k, lane),
                  ldb_pk(pkWe, DIM, k, col0, lane), acc);
    int half = lane >> 4, c = lane & 15;
    float bias = be[col0 + c];
#pragma unroll
    for (int r = 0; r < 8; r++)
      ef[(r + 8 * half) * LS + col0 + c] = acc[r] + bias;
  }
  __syncthreads();

  int h = wave;
  for (int row = 0; row < 16; row++) {
    int e = e0 + row;
    if (e >= E) break;
    int rc = receivers[e];
    float coef = wlog[(size_t)h * E + e] / segsum[rc * NHEAD + h];
    float* dst = agg + (size_t)rc * HD + h * DIM;
#pragma unroll
    for (int j = 0; j < 4; j++) {
      int col = lane + 32 * j;
      atomicAdd(&dst[col], coef * (sa[row * LS + col] + ef[row * LS + col]));
    }
  }
}

// ---------------- out = relu(x + agg @ W_out + b_out)  (or head-mean) ----------------
__global__ void __launch_bounds__(32)
k_final(const float* __restrict__ x, const float* __restrict__ agg,
        const float2* __restrict__ pkWo, const float* __restrict__ bo,
        const int* __restrict__ avg_flag, float* __restrict__ out, int N) {
  int tile = blockIdx.x;
  int tm = tile >> 3, tn = tile & 7;
  int lane = threadIdx.x;
  int row0 = tm * 16, col0 = tn * 16;
  if (row0 >= N) return;
  int half = lane >> 4, c = lane & 15;
  if (avg_flag[0] == 0) {
    v8f acc = {};
#pragma unroll 8
    for (int k = 0; k < HD; k += 4)
      acc = wmma4(lda_frag(agg, HD, row0, k, lane),
                  ldb_pk(pkWo, DIM, k, col0, lane), acc);
    float bias = bo[col0 + c];
#pragma unroll
    for (int r = 0; r < 8; r++) {
      size_t idx = (size_t)(row0 + r + 8 * half) * DIM + col0 + c;
      float v = x[idx] + acc[r] + bias;
      out[idx] = v > 0.f ? v : 0.f;
    }
  } else {
#pragma unroll
    for (int r = 0; r < 8; r++) {
      int row = row0 + r + 8 * half;
      int col = col0 + c;
      float s = 0.f;
#pragma unroll
      for (int hh = 0; hh < NHEAD; hh++)
        s += agg[(size_t)row * HD + hh * DIM + col];
      size_t idx = (size_t)row * DIM + col;
      float v = x[idx] + s * 0.25f;
      out[idx] = v > 0.f ? v : 0.f;
    }
  }
}

extern "C" void kernel_launch(void* const* d_in, const int* in_sizes, int n_in,
                              void* d_out, int out_size, void* d_ws,
                              size_t ws_size, hipStream_t stream) {
  const float* nodes = (const float*)d_in[0];
  const float* edges = (const float*)d_in[1];
  const int* senders = (const int*)d_in[2];
  const int* receivers = (const int*)d_in[3];
  const int* avg_flag = (const int*)d_in[4];
  const float* Wn = (const float*)d_in[5];
  const float* bn = (const float*)d_in[6];
  const float* We = (const float*)d_in[7];
  const float* be = (const float*)d_in[8];
  const float* Wks = (const float*)d_in[9];
  const float* bks = (const float*)d_in[10];
  const float* Wkr = (const float*)d_in[11];
  const float* bkr = (const float*)d_in[12];
  const float* Wo = (const float*)d_in[13];
  const float* bo = (const float*)d_in[14];
  float* out = (float*)d_out;

  int N = in_sizes[0] / FN; // 50000
  int E = in_sizes[2];      // 800000

  // Workspace (floats): x | wlog | segmax | segsum | agg | packed weights
  float* ws = (float*)d_ws;
  float* x = ws;
  float* wlog = x + (size_t)N * DIM;
  float* segmax = wlog + (size_t)NHEAD * E;
  float* segsum = segmax + (size_t)N * NHEAD;
  float* agg = segsum + (size_t)N * NHEAD;
  float* pk = agg + (size_t)N * HD;
  float2* pkWn = (float2*)pk;                       // FN*DIM floats
  float2* pkWe = pkWn + (size_t)FN * DIM / 2;       // FE*DIM floats
  float2* pkWks = pkWe + (size_t)FE * DIM / 2;      // NHEAD*DIM*DIM floats
  float2* pkWkr = pkWks + (size_t)NHEAD * DIM * DIM / 2;
  float2* pkWo = pkWkr + (size_t)NHEAD * DIM * DIM / 2; // HD*DIM floats

  // Repack weights into B-fragment order (one b64 per fragment).
  k_pack<<<(FN * DIM / 2 + 255) / 256, 256, 0, stream>>>(Wn, pkWn, FN, DIM);
  k_pack<<<(FE * DIM / 2 + 255) / 256, 256, 0, stream>>>(We, pkWe, FE, DIM);
  k_pack<<<(NHEAD * DIM * DIM / 2 + 255) / 256, 256, 0, stream>>>(
      Wks, pkWks, NHEAD * DIM, DIM);
  k_pack<<<(NHEAD * DIM * DIM / 2 + 255) / 256, 256, 0, stream>>>(
      Wkr, pkWkr, NHEAD * DIM, DIM);
  k_pack<<<(HD * DIM / 2 + 255) / 256, 256, 0, stream>>>(Wo, pkWo, HD, DIM);

  int NH = N * NHEAD, NHD = N * HD;
  k_init<<<(NHD + 255) / 256, 256, 0, stream>>>(segmax, segsum, agg, NH, NHD);

  int row_tiles = (N + 15) / 16;
  k_proj_nodes<<<row_tiles * (DIM / 16), 32, 0, stream>>>(nodes, pkWn, bn, x, N);

  int etiles = (E + 15) / 16;
  k_logits<<<etiles, 128, 0, stream>>>(x, edges, senders, receivers, pkWe, be,
                                       pkWks, bks, pkWkr, bkr, wlog, segmax, E);

  dim3 g2((E + 255) / 256, NHEAD);
  k_expsum<<<g2, 256, 0, stream>>>(segmax, receivers, wlog, segsum, E);

  k_msg<<<etiles, 128, 0, stream>>>(x, edges, senders, receivers, pkWe, be,
                                    wlog, segsum, agg, E);

  k_final<<<row_tiles * (DIM / 16), 32, 0, stream>>>(x, agg, pkWo, bo, avg_flag,
                                                     out, N);
}